// MambaBlock_82669530513866
// MI455X (gfx1250) — compile-verified
//
#include <hip/hip_runtime.h>
#include <math.h>
#include <stdint.h>

// ---------------------------------------------------------------------------
// MambaBlock for MI455X (gfx1250), fp32 via V_WMMA_F32_16X16X4_F32, with
// double-buffered GLOBAL_LOAD_ASYNC_TO_LDS_B128 staging (ASYNCcnt pipeline).
//
// Pipeline:
//   1. proj  = x @ W1^T + b1                      (WMMA GEMM, bias epilogue)
//   2. xn    = LN(proj[:, :1024])                 (fused reduce)
//   3. xc    = SiLU(conv3(xn) + cb)               (3 accumulated WMMA GEMMs)
//   4. xb    = xc @ B                             (WMMA GEMM)
//   5. h_t   = h_{t-1} @ A + xb_t                 (512 sequential WMMA steps,
//                                                  h async-staged in LDS)
//   6. ys    = H @ C                              (WMMA GEMM)
//   7. yn    = LN(ys * SiLU(res))                 (fused)
//   8. out   = yn @ Wo^T + ob + xc                (WMMA GEMM, bias+resid)
// ---------------------------------------------------------------------------

typedef __attribute__((ext_vector_type(2))) float v2f;
typedef __attribute__((ext_vector_type(8))) float v8f;

#define DM    512
#define DI    1024
#define LSEQ  512
#define NB    16
#define NTOK  (NB * LSEQ)          // 8192

// LDS tile strides (floats). Multiples of 4 -> async b128 dsts are 16B
// aligned. 36: 36*dm % 64 != 0 for dm in 1..15 and always a multiple of 4,
// so A-frag reads (16 lanes, stride 36, +2 cross-half) are conflict-free.
// 72: cross-half bank distance is 16+delta, delta in [-15,15] -> never 0.
#define SA 36
#define SB 72

// D = A(16x4) * B(4x16) + C, fp32.
__device__ __forceinline__ v8f wmma4(v2f a, v2f b, v8f c) {
  return __builtin_amdgcn_wmma_f32_16x16x4_f32(
      /*neg_a=*/false, a, /*neg_b=*/false, b,
      /*c_mod=*/(short)0, c, /*reuse_a=*/false, /*reuse_b=*/false);
}

// Async copy of 16 bytes from global to LDS (per active lane). LDS dst is the
// 32-bit AS(3) offset = low 32 bits of the flat shared pointer.
__device__ __forceinline__ void async16(float* lds_dst, const float* gsrc) {
  const uint32_t loff = (uint32_t)(uintptr_t)lds_dst;
  asm volatile("global_load_async_to_lds_b128 %0, %1, off"
               :: "v"(loff), "v"(gsrc) : "memory");
}
__device__ __forceinline__ void wait_async0() {
  asm volatile("s_wait_asynccnt 0x0" ::: "memory");
}

// ---------------------------------------------------------------------------
// Generic WMMA GEMM: C[MxN] = A[MxK] @ B[KxN] (+bias[col]) (+resid).
// Block tile 64x64, 16 waves (one 16x16 WMMA tile each). K staged 32 at a
// time through double-buffered LDS via async copies: issue tile s+1, compute
// tile s, s_wait_asynccnt 0, one barrier. M,N mult of 64; K mult of 32.
// EPI: 0 = none, 1 = +bias, 3 = +bias +resid
// ---------------------------------------------------------------------------
template <int EPI>
__global__ __launch_bounds__(512) void gemm64(
    const float* __restrict__ A, const float* __restrict__ B,
    float* __restrict__ C, int M, int N, int K,
    const float* __restrict__ bias, const float* __restrict__ resid) {
  __shared__ float As[2][64 * SA];
  __shared__ float Bs[2][32 * SB];

  const int tid  = threadIdx.x;
  const int lane = tid & 31;
  const int wave = tid >> 5;
  const int wy   = wave >> 2, wx = wave & 3;
  const int half = lane >> 4;     // ISA 7.12.2 lane-half split
  const int l15  = lane & 15;
  const int m0   = blockIdx.y * 64;
  const int n0   = blockIdx.x * 64;

  const int ar = tid >> 3,  ac = (tid & 7) << 2;   // A stage: 64 rows x 32
  const int br = tid >> 4,  bc = (tid & 15) << 2;  // B stage: 32 rows x 64

  const float* Ag = A + (size_t)(m0 + ar) * K + ac;
  const float* Bg = B + (size_t)br * N + n0 + bc;

  auto stage = [&](int k0, int p) {
    async16(&As[p][ar * SA + ac], Ag + k0);
    async16(&Bs[p][br * SB + bc], Bg + (size_t)k0 * N);
  };

  stage(0, 0);
  wait_async0();
  __syncthreads();

  v8f c = {0.f, 0.f, 0.f, 0.f, 0.f, 0.f, 0.f, 0.f};
  const int steps = K >> 5;
  for (int s = 0; s < steps; ++s) {
    const int p = s & 1;
    if (s + 1 < steps) stage((s + 1) << 5, p ^ 1);   // overlap with compute

    const float* Ap = &As[p][(wy * 16 + l15) * SA];
    const float* Bp = &Bs[p][wx * 16 + l15];
#pragma unroll
    for (int kk = 0; kk < 32; kk += 4) {
      const int ka = kk + half * 2;
      v2f a;  a.x = Ap[ka];  a.y = Ap[ka + 1];
      const float* bp = Bp + ka * SB;
      v2f b;  b.x = bp[0];  b.y = bp[SB];
      c = wmma4(a, b, c);
    }
    wait_async0();      // next tile landed
    __syncthreads();    // everyone done with buffer p
  }

#pragma unroll
  for (int r = 0; r < 8; ++r) {   // D layout: VGPR r -> M = r (+8 for half 1)
    const int row = m0 + wy * 16 + r + half * 8;
    const int col = n0 + wx * 16 + l15;
    float v = c[r];
    if (EPI >= 1) v += bias[col];
    if (EPI == 3) v += resid[(size_t)row * N + col];
    C[(size_t)row * N + col] = v;
  }
}

// ---------------------------------------------------------------------------
// Conv as 3 accumulated GEMMs with per-row shift (zero-padded at sequence
// boundaries, never crossing batch boundaries), fused bias + SiLU epilogue.
// Same double-buffered async pipeline; zero-pad rows use plain LDS stores
// (DScnt) which the same barrier flushes.
// Wk: 3 planes of [i][o] (pre-rearranged from conv_w[o][i][k]).
// ---------------------------------------------------------------------------
__global__ __launch_bounds__(512) void conv_gemm_silu(
    const float* __restrict__ xn, const float* __restrict__ Wk,
    const float* __restrict__ cb, float* __restrict__ xc) {
  __shared__ float As[2][64 * SA];
  __shared__ float Bs[2][32 * SB];

  const int tid  = threadIdx.x;
  const int lane = tid & 31;
  const int wave = tid >> 5;
  const int wy   = wave >> 2, wx = wave & 3;
  const int half = lane >> 4;
  const int l15  = lane & 15;
  const int m0   = blockIdx.y * 64;
  const int n0   = blockIdx.x * 64;

  const int ar = tid >> 3,  ac = (tid & 7) << 2;
  const int br = tid >> 4,  bc = (tid & 15) << 2;
  const int nrow = m0 + ar;
  const int trow = nrow & (LSEQ - 1);

  // it = 0..95 : tap k = it>>5, k-chunk i0 = (it&31)*32
  auto stage = [&](int it, int p) {
    const int k  = it >> 5;
    const int i0 = (it & 31) << 5;
    const int shift = k - 1;
    float* ad = &As[p][ar * SA + ac];
    if ((unsigned)(trow + shift) < (unsigned)LSEQ) {
      async16(ad, xn + (size_t)(nrow + shift) * DI + i0 + ac);
    } else {
      ad[0] = 0.f; ad[1] = 0.f; ad[2] = 0.f; ad[3] = 0.f;
    }
    async16(&Bs[p][br * SB + bc],
            Wk + (size_t)k * DI * DI + (size_t)(i0 + br) * DI + n0 + bc);
  };

  stage(0, 0);
  wait_async0();
  __syncthreads();

  v8f c = {0.f, 0.f, 0.f, 0.f, 0.f, 0.f, 0.f, 0.f};
  for (int it = 0; it < 96; ++it) {
    const int p = it & 1;
    if (it + 1 < 96) stage(it + 1, p ^ 1);

    const float* Ap = &As[p][(wy * 16 + l15) * SA];
    const float* Bp = &Bs[p][wx * 16 + l15];
#pragma unroll
    for (int kk = 0; kk < 32; kk += 4) {
      const int ka = kk + half * 2;
      v2f a;  a.x = Ap[ka];  a.y = Ap[ka + 1];
      const float* bp = Bp + ka * SB;
      v2f b;  b.x = bp[0];  b.y = bp[SB];
      c = wmma4(a, b, c);
    }
    wait_async0();
    __syncthreads();
  }

#pragma unroll
  for (int r = 0; r < 8; ++r) {
    const int row = m0 + wy * 16 + r + half * 8;
    const int col = n0 + wx * 16 + l15;
    float v = c[r] + cb[col];
    v = v / (1.f + expf(-v));     // SiLU
    xc[(size_t)row * DI + col] = v;
  }
}

// ---------------------------------------------------------------------------
// One scan step: Hcur(16x1024) = Hprev(16x1024) @ A(1024x1024) + xb_t.
// Hprev async-staged into LDS (64KB) with XOR swizzle: A-frag reads are 16
// lanes, same k, stride-1024 rows -> swizzle k^(4m) spreads them over 16
// distinct banks (and the +2 cross-half offset cannot collide: 4m XOR only
// touches bits >= 2). Swizzle preserves 16B alignment for async b128.
// Grid: 16 blocks x 128 threads; each wave owns 16 output cols, M=16=batch.
// ---------------------------------------------------------------------------
__device__ __forceinline__ int hswz(int m, int kk) {
  return m * DI + (kk ^ ((m & 15) << 2));
}

__global__ __launch_bounds__(128) void scan_step(
    const float* __restrict__ Hprev, float* __restrict__ Hcur,
    const float* __restrict__ xbt, const float* __restrict__ Amat) {
  __shared__ float Hs[NB * DI];   // 64 KB

  const int tid  = threadIdx.x;
  const int lane = tid & 31;
  const int wave = tid >> 5;
  const int half = lane >> 4;
  const int l15  = lane & 15;
  const int col  = blockIdx.x * 64 + wave * 16 + l15;

  for (int i = tid * 4; i < NB * DI; i += 128 * 4)
    async16(&Hs[hswz(i >> 10, i & (DI - 1))], Hprev + i);

  v8f c;
#pragma unroll
  for (int r = 0; r < 8; ++r)     // init from xb_t: batch = r + 8*half
    c[r] = xbt[(size_t)(r + half * 8) * (LSEQ * DI) + col];

  wait_async0();
  __syncthreads();

#pragma unroll 4
  for (int k0 = 0; k0 < DI; k0 += 4) {
    const int ka = k0 + half * 2;
    v2f a;
    a.x = Hs[hswz(l15, ka)];
    a.y = Hs[hswz(l15, ka) + 1];  // ka+1 only flips bit0: same swizzle base
    const float* bp = Amat + (size_t)ka * DI + col;
    v2f b;  b.x = bp[0];  b.y = bp[DI];
    c = wmma4(a, b, c);
  }

#pragma unroll
  for (int r = 0; r < 8; ++r)
    Hcur[(size_t)(r + half * 8) * DI + col] = c[r];
}

// ---------------------------------------------------------------------------
// Elementwise / reduction kernels
// ---------------------------------------------------------------------------
__device__ __forceinline__ void blockReduce2(float& a, float& b) {
  __shared__ float sa[8], sb[8];
#pragma unroll
  for (int off = 16; off > 0; off >>= 1) {   // wave32 reduce
    a += __shfl_xor(a, off, 32);
    b += __shfl_xor(b, off, 32);
  }
  const int lane = threadIdx.x & 31, w = threadIdx.x >> 5;
  if (lane == 0) { sa[w] = a; sb[w] = b; }
  __syncthreads();
  a = 0.f; b = 0.f;
#pragma unroll
  for (int i = 0; i < 8; ++i) { a += sa[i]; b += sb[i]; }
}

__global__ __launch_bounds__(256) void layernorm_rows(
    const float* __restrict__ in, int in_stride,
    const float* __restrict__ g, const float* __restrict__ b,
    float* __restrict__ out) {
  const int n = blockIdx.x;
  const float* row = in + (size_t)n * in_stride;
  float v[4], s = 0.f, ss = 0.f;
#pragma unroll
  for (int j = 0; j < 4; ++j) {
    const float x = row[threadIdx.x + j * 256];
    v[j] = x;  s += x;  ss += x * x;
  }
  blockReduce2(s, ss);
  const float mu = s * (1.f / DI);
  const float rs = rsqrtf(ss * (1.f / DI) - mu * mu + 1e-5f);
#pragma unroll
  for (int j = 0; j < 4; ++j) {
    const int idx = threadIdx.x + j * 256;
    out[(size_t)n * DI + idx] = (v[j] - mu) * rs * g[idx] + b[idx];
  }
}

// yn[n] = LN( ys[(t*16+b)] * SiLU(res[n]) )  with n = b*LSEQ + t
__global__ __launch_bounds__(256) void mul_silu_ln(
    const float* __restrict__ ys, const float* __restrict__ proj,
    const float* __restrict__ g, const float* __restrict__ b,
    float* __restrict__ out) {
  const int n = blockIdx.x;
  const int bb = n >> 9, t = n & (LSEQ - 1);
  const float* yrow = ys + (size_t)(t * NB + bb) * DI;
  const float* rrow = proj + (size_t)n * (2 * DI) + DI;
  float v[4], s = 0.f, ss = 0.f;
#pragma unroll
  for (int j = 0; j < 4; ++j) {
    const int idx = threadIdx.x + j * 256;
    const float r = rrow[idx];
    const float x = yrow[idx] * (r / (1.f + expf(-r)));
    v[j] = x;  s += x;  ss += x * x;
  }
  blockReduce2(s, ss);
  const float mu = s * (1.f / DI);
  const float rs = rsqrtf(ss * (1.f / DI) - mu * mu + 1e-5f);
#pragma unroll
  for (int j = 0; j < 4; ++j) {
    const int idx = threadIdx.x + j * 256;
    out[(size_t)n * DI + idx] = (v[j] - mu) * rs * g[idx] + b[idx];
  }
}

// out[k][n] = in[n][k]; in is R x C, out is C x R. R,C multiples of 32.
__global__ __launch_bounds__(256) void transpose_k(
    const float* __restrict__ in, float* __restrict__ out, int R, int C) {
  __shared__ float tile[32][33];
  const int c0 = blockIdx.x * 32, r0 = blockIdx.y * 32;
  const int tx = threadIdx.x & 31, ty = threadIdx.x >> 5;
#pragma unroll
  for (int j = 0; j < 32; j += 8)
    tile[ty + j][tx] = in[(size_t)(r0 + ty + j) * C + c0 + tx];
  __syncthreads();
#pragma unroll
  for (int j = 0; j < 32; j += 8)
    out[(size_t)(c0 + ty + j) * R + r0 + tx] = tile[tx][ty + j];
}

// conv_w[o][i][k] (DIxDIx3) -> 3 planes Wk[k][i][o]
__global__ __launch_bounds__(256) void conv_w_rearrange(
    const float* __restrict__ w, float* __restrict__ Wk) {
  const int idx = blockIdx.x * 256 + threadIdx.x;  // i*DI + o
  const int i = idx >> 10, o = idx & (DI - 1);
#pragma unroll
  for (int k = 0; k < 3; ++k)
    Wk[(size_t)k * DI * DI + idx] = w[(size_t)o * (DI * 3) + i * 3 + k];
}

__global__ void zero_f(float* p, int n) {
  const int i = blockIdx.x * 256 + threadIdx.x;
  if (i < n) p[i] = 0.f;
}

// ---------------------------------------------------------------------------
extern "C" void kernel_launch(void* const* d_in, const int* in_sizes, int n_in,
                              void* d_out, int out_size, void* d_ws,
                              size_t ws_size, hipStream_t stream) {
  (void)in_sizes; (void)n_in; (void)out_size; (void)ws_size;
  const float* x   = (const float*)d_in[0];
  const float* w1  = (const float*)d_in[1];   // (2048, 512)
  const float* b1  = (const float*)d_in[2];
  const float* g1  = (const float*)d_in[3];
  const float* be1 = (const float*)d_in[4];
  const float* cw  = (const float*)d_in[5];   // (1024, 1024, 3)
  const float* cb  = (const float*)d_in[6];
  const float* Am  = (const float*)d_in[7];   // (1024, 1024)
  const float* Bm  = (const float*)d_in[8];
  const float* Cm  = (const float*)d_in[9];
  const float* g2  = (const float*)d_in[10];
  const float* be2 = (const float*)d_in[11];
  const float* ow  = (const float*)d_in[12];  // (1024, 1024)
  const float* ob  = (const float*)d_in[13];
  float* out = (float*)d_out;

  float* ws = (float*)d_ws;
  size_t off = 0;
  auto alloc = [&](size_t n) { float* p = ws + off; off += n; return p; };
  float* proj = alloc((size_t)NTOK * 2 * DI);      // 8192 x 2048
  float* xn   = alloc((size_t)NTOK * DI);          // reused as yn
  float* xc   = alloc((size_t)NTOK * DI);
  float* xb   = alloc((size_t)NTOK * DI);          // reused as ys
  float* Hbuf = alloc((size_t)(LSEQ + 1) * NB * DI);  // slot 0 = h0 = 0
  float* Wt1  = alloc((size_t)DM * 2 * DI);        // in_proj_w^T
  float* Wk3  = alloc((size_t)3 * DI * DI);        // conv taps as [i][o]
  float* Wt2  = alloc((size_t)DI * DI);            // out_w^T

  // --- weight prep (one-shot, L2-resident afterwards) ---
  transpose_k<<<dim3(DM / 32, (2 * DI) / 32), 256, 0, stream>>>(w1, Wt1, 2 * DI, DM);
  transpose_k<<<dim3(DI / 32, DI / 32), 256, 0, stream>>>(ow, Wt2, DI, DI);
  conv_w_rearrange<<<(DI * DI) / 256, 256, 0, stream>>>(cw, Wk3);

  // --- 1. in_proj ---
  gemm64<1><<<dim3((2 * DI) / 64, NTOK / 64), 512, 0, stream>>>(
      x, Wt1, proj, NTOK, 2 * DI, DM, b1, nullptr);
  // --- 2. LN1 on xs = proj[:, :1024] ---
  layernorm_rows<<<NTOK, 256, 0, stream>>>(proj, 2 * DI, g1, be1, xn);
  // --- 3. conv + SiLU ---
  conv_gemm_silu<<<dim3(DI / 64, NTOK / 64), 512, 0, stream>>>(xn, Wk3, cb, xc);
  // --- 4. xb = xc @ B (hoists x@B out of the recurrence) ---
  gemm64<0><<<dim3(DI / 64, NTOK / 64), 512, 0, stream>>>(
      xc, Bm, xb, NTOK, DI, DI, nullptr, nullptr);
  // --- 5. sequential scan (latency chain; 512 graph-captured launches) ---
  zero_f<<<(NB * DI) / 256, 256, 0, stream>>>(Hbuf, NB * DI);
  const float* hp = Hbuf;
  for (int t = 0; t < LSEQ; ++t) {
    float* hc = Hbuf + (size_t)(t + 1) * NB * DI;
    scan_step<<<16, 128, 0, stream>>>(hp, hc, xb + (size_t)t * DI, Am);
    hp = hc;
  }
  // --- 6. ys = H @ C (rows in (t,b) order) ---
  float* ys = xb;
  gemm64<0><<<dim3(DI / 64, NTOK / 64), 512, 0, stream>>>(
      Hbuf + NB * DI, Cm, ys, NTOK, DI, DI, nullptr, nullptr);
  // --- 7. yn = LN(ys * SiLU(res)) ---
  float* yn = xn;
  mul_silu_ln<<<NTOK, 256, 0, stream>>>(ys, proj, g2, be2, yn);
  // --- 8. out = yn @ out_w^T + ob + xc ---
  gemm64<3><<<dim3(DI / 64, NTOK / 64), 512, 0, stream>>>(
      yn, Wt2, out, NTOK, DI, DI, ob, xc);
}